// GCNBase_5111011083135
// MI455X (gfx1250) — compile-verified
//
#include <hip/hip_runtime.h>
#include <math.h>

typedef float v2f __attribute__((ext_vector_type(2)));
typedef float v8f __attribute__((ext_vector_type(8)));

// Relaxed agent-scope FP32 atomic add -> global_atomic_add_f32 on gfx1250.
__device__ inline void atomicAddF(float* p, float v) {
    __hip_atomic_fetch_add(p, v, __ATOMIC_RELAXED, __HIP_MEMORY_SCOPE_AGENT);
}

// ---------------------------------------------------------------------------
// Degree accumulation: deg[dst] += ew  (reference adds +1.0 self-loop later)
// ---------------------------------------------------------------------------
__global__ void deg_kernel(const int* __restrict__ dst, const float* __restrict__ ew,
                           float* __restrict__ deg, int E) {
    int e = blockIdx.x * blockDim.x + threadIdx.x;
    if (e < E) atomicAddF(deg + dst[e], ew[e]);
}

__global__ void dinv_kernel(const float* __restrict__ deg, float* __restrict__ dinv, int N) {
    int i = blockIdx.x * blockDim.x + threadIdx.x;
    if (i < N) dinv[i] = rsqrtf(deg[i] + 1.0f);   // deg+1 > 0 always
}

// ---------------------------------------------------------------------------
// fp32 WMMA GEMM: C[M x NC] = A[M x K] * W[K x NC], K compile-time (128).
// Block = NWAVES waves; blockIdx.x -> 16-row stripe, blockIdx.y -> group of
// NWAVES 16-col tiles.  A tile (16xK) and a transposed W slice (NWAVES*16 x K,
// zero-padded past NC) are staged in LDS with stride K+2 (8B-aligned, no bank
// conflicts), so the unrolled inner loop is ds_load_b64 x2 + v_wmma only.
// V_WMMA_F32_16X16X4_F32: A frag lanes 0-15 K{0,1}, 16-31 K{2,3};
// C/D frag VGPR r -> row r (lanes 0-15) / row r+8 (lanes 16-31).
// ---------------------------------------------------------------------------
template <int K, int NC, int NWAVES>
__global__ void gemm_wmma_f32(const float* __restrict__ A, const float* __restrict__ W,
                              float* __restrict__ C, int M) {
    constexpr int NCPAD = NWAVES * 16;   // staged cols per block
    constexpr int LSTR  = K + 2;         // padded LDS stride (even -> b64 aligned)
    __shared__ float At[16 * LSTR];
    __shared__ float Wt[NCPAD * LSTR];

    const int tid  = threadIdx.x;
    const int lane = tid & 31;
    const int wave = tid >> 5;
    const int m0   = blockIdx.x * 16;
    const int c0   = blockIdx.y * NCPAD;

    // ---- stage A tile (coalesced rows) ----
    for (int idx = tid; idx < 16 * K; idx += NWAVES * 32) {
        int r = idx / K, k = idx % K;
        int row = m0 + r; if (row >= M) row = M - 1;
        At[r * LSTR + k] = A[(size_t)row * K + k];
    }
    // ---- stage W slice transposed, zero-padded past NC ----
    for (int idx = tid; idx < K * NCPAD; idx += NWAVES * 32) {
        int k = idx / NCPAD, c = idx % NCPAD;
        int col = c0 + c;
        Wt[c * LSTR + k] = (col < NC) ? W[(size_t)k * NC + col] : 0.0f;
    }
    __syncthreads();

    const int half = lane >> 4;          // 0 -> K{0,1}, 1 -> K{2,3}
    const int l15  = lane & 15;
    const float* ap = At + l15 * LSTR + half * 2;
    const float* bp = Wt + (wave * 16 + l15) * LSTR + half * 2;

    v8f acc = {};
#pragma unroll
    for (int k = 0; k < K; k += 4) {
        v2f a = *reinterpret_cast<const v2f*>(ap + k);   // ds_load_b64
        v2f b = *reinterpret_cast<const v2f*>(bp + k);   // ds_load_b64
        // (neg_a, A, neg_b, B, c_mod, C, reuse_a, reuse_b)
        acc = __builtin_amdgcn_wmma_f32_16x16x4_f32(false, a, false, b,
                                                    (short)0, acc, false, false);
    }

    const int bcol = c0 + wave * 16 + l15;
    if (bcol < NC) {
#pragma unroll
        for (int r = 0; r < 8; ++r) {
            int row = m0 + r + half * 8;
            if (row < M) C[(size_t)row * NC + bcol] = acc[r];
        }
    }
}

// ---------------------------------------------------------------------------
// agg = xw * dinv^2 + bias   (self-loop message + bias; initializes agg)
// ---------------------------------------------------------------------------
__global__ void selfloop_bias(const float* __restrict__ xw, const float* __restrict__ dinv,
                              const float* __restrict__ bias, float* __restrict__ agg,
                              int N, int D) {
    size_t i = (size_t)blockIdx.x * blockDim.x + threadIdx.x;
    size_t total = (size_t)N * D;
    if (i >= total) return;
    int node = (int)(i / D);
    int c    = (int)(i % D);
    float di = dinv[node];
    agg[i] = xw[i] * (di * di) + bias[c];
}

// ---------------------------------------------------------------------------
// Edge scatter, D=128: one wave per edge; lane moves 4 channels (float4).
// xw (51MB) fits in the 192MB L2 -> gathers & atomics are L2-resident.
// ---------------------------------------------------------------------------
__global__ void scatter_edges128(const float* __restrict__ xw, const int* __restrict__ src,
                                 const int* __restrict__ dst, const float* __restrict__ ew,
                                 const float* __restrict__ dinv, float* __restrict__ agg, int E) {
    int e    = (int)(((size_t)blockIdx.x * blockDim.x + threadIdx.x) >> 5);
    int lane = threadIdx.x & 31;
    if (e >= E) return;
    int s = src[e], d = dst[e];
    float nrm = dinv[s] * ew[e] * dinv[d];
    float4 v = reinterpret_cast<const float4*>(xw + (size_t)s * 128)[lane];
    float* out = agg + (size_t)d * 128 + lane * 4;
    atomicAddF(out + 0, v.x * nrm);
    atomicAddF(out + 1, v.y * nrm);
    atomicAddF(out + 2, v.z * nrm);
    atomicAddF(out + 3, v.w * nrm);
}

// Edge scatter, D=40: lane covers col `lane`, lanes 0-7 also col lane+32.
__global__ void scatter_edges40(const float* __restrict__ xw, const int* __restrict__ src,
                                const int* __restrict__ dst, const float* __restrict__ ew,
                                const float* __restrict__ dinv, float* __restrict__ agg, int E) {
    int e    = (int)(((size_t)blockIdx.x * blockDim.x + threadIdx.x) >> 5);
    int lane = threadIdx.x & 31;
    if (e >= E) return;
    int s = src[e], d = dst[e];
    float nrm = dinv[s] * ew[e] * dinv[d];
    const float* xr = xw + (size_t)s * 40;
    float* out = agg + (size_t)d * 40;
    atomicAddF(out + lane, xr[lane] * nrm);
    if (lane < 8) atomicAddF(out + 32 + lane, xr[32 + lane] * nrm);
}

// ---------------------------------------------------------------------------
// BatchNorm stats: register accumulation over a 256-row stripe, LDS combine,
// then one global float-atomic per column per block.  D fixed at 128.
// ---------------------------------------------------------------------------
__global__ void bn_stats(const float* __restrict__ h, float* __restrict__ colsum,
                         float* __restrict__ colsq, int N) {
    __shared__ float ssum[256];
    __shared__ float ssq[256];
    int col  = threadIdx.x & 127;
    int half = threadIdx.x >> 7;
    int r    = blockIdx.x * 256 + half;
    int rend = min(N, blockIdx.x * 256 + 256);
    float s = 0.0f, q = 0.0f;
    for (; r < rend; r += 2) {
        float v = h[(size_t)r * 128 + col];
        s += v; q += v * v;
    }
    ssum[threadIdx.x] = s;
    ssq[threadIdx.x]  = q;
    __syncthreads();
    if (half == 0) {
        s += ssum[threadIdx.x + 128];
        q += ssq[threadIdx.x + 128];
        atomicAddF(colsum + col, s);
        atomicAddF(colsq + col, q);
    }
}

__global__ void bn_finalize(const float* __restrict__ colsum, const float* __restrict__ colsq,
                            const float* __restrict__ gamma, const float* __restrict__ beta,
                            float* __restrict__ scsh, int N) {
    int c = threadIdx.x;                       // 128 threads
    float inv_n = 1.0f / (float)N;
    float mean  = colsum[c] * inv_n;
    float var   = colsq[c] * inv_n - mean * mean;
    float sc    = gamma[c] * rsqrtf(var + 1e-5f);
    scsh[c]       = sc;
    scsh[128 + c] = beta[c] - mean * sc;
}

__global__ void bn_apply_relu(float* __restrict__ h, const float* __restrict__ scsh,
                              size_t total) {
    size_t i = (size_t)blockIdx.x * blockDim.x + threadIdx.x;
    if (i >= total) return;
    int c = (int)(i & 127);
    float v = h[i] * scsh[c] + scsh[128 + c];
    h[i] = v > 0.0f ? v : 0.0f;
}

// ---------------------------------------------------------------------------
// Row-wise log_softmax over 40 classes: wave per node, shfl_xor reductions.
// ---------------------------------------------------------------------------
__global__ void log_softmax40(const float* __restrict__ in, float* __restrict__ out, int N) {
    int node = (int)(((size_t)blockIdx.x * blockDim.x + threadIdx.x) >> 5);
    int lane = threadIdx.x & 31;
    if (node >= N) return;
    const float* row = in + (size_t)node * 40;
    float v0 = row[lane];
    float v1 = (lane < 8) ? row[32 + lane] : -__builtin_inff();
    float m = fmaxf(v0, v1);
#pragma unroll
    for (int off = 16; off > 0; off >>= 1) m = fmaxf(m, __shfl_xor(m, off, 32));
    float s = __expf(v0 - m) + ((lane < 8) ? __expf(v1 - m) : 0.0f);
#pragma unroll
    for (int off = 16; off > 0; off >>= 1) s += __shfl_xor(s, off, 32);
    float lse = m + __logf(s);
    float* orow = out + (size_t)node * 40;
    orow[lane] = v0 - lse;
    if (lane < 8) orow[32 + lane] = v1 - lse;
}

// ---------------------------------------------------------------------------
// Host orchestration (graph-capture safe: only kernels + hipMemsetAsync)
// ---------------------------------------------------------------------------
extern "C" void kernel_launch(void* const* d_in, const int* in_sizes, int n_in,
                              void* d_out, int out_size, void* d_ws, size_t ws_size,
                              hipStream_t stream) {
    const float* x   = (const float*)d_in[0];
    const int*   ei  = (const int*)d_in[1];     // edge_index [2, E] (int32 under JAX defaults)
    const float* ew  = (const float*)d_in[2];
    const float* W0  = (const float*)d_in[3];
    const float* b0  = (const float*)d_in[4];
    const float* g0  = (const float*)d_in[5];
    const float* be0 = (const float*)d_in[6];
    const float* W1  = (const float*)d_in[7];
    const float* b1  = (const float*)d_in[8];
    const float* g1  = (const float*)d_in[9];
    const float* be1 = (const float*)d_in[10];
    const float* W2  = (const float*)d_in[11];
    const float* b2  = (const float*)d_in[12];

    const int D  = 128, DOUT = 40;
    const int N  = in_sizes[0] / D;
    const int E  = in_sizes[1] / 2;
    const int* src = ei;
    const int* dst = ei + E;
    float* out = (float*)d_out;

    // workspace carve-out
    char* ws = (char*)d_ws;
    size_t off = 0;
    auto carve = [&](size_t bytes) -> float* {
        float* p = (float*)(ws + off);
        off = (off + bytes + 255) & ~(size_t)255;
        return p;
    };
    float* bufA   = carve((size_t)N * D * sizeof(float));   // xw / agg2
    float* bufB   = carve((size_t)N * D * sizeof(float));   // agg / h
    float* deg    = carve((size_t)N * sizeof(float));
    float* dinv   = carve((size_t)N * sizeof(float));
    float* colsum = carve(128 * sizeof(float));
    float* colsq  = carve(128 * sizeof(float));
    float* scsh   = carve(256 * sizeof(float));
    (void)ws_size; (void)n_in; (void)out_size;

    const int rowTiles   = (N + 15) / 16;
    const size_t totND   = (size_t)N * D;
    const size_t totNDo  = (size_t)N * DOUT;
    const int edgeBlocks = (E + 7) / 8;           // 8 waves/block, wave per edge
    const int statBlocks = (N + 255) / 256;
    const dim3 gemmGrid128(rowTiles, 2);          // 2 x (4 waves x 16 cols) = 128 cols
    const dim3 gemmGrid40(rowTiles, 1);           // 3 waves x 16 cols >= 40 cols

    // degrees + symmetric normalization
    hipMemsetAsync(deg, 0, (size_t)N * sizeof(float), stream);
    deg_kernel<<<(E + 255) / 256, 256, 0, stream>>>(dst, ew, deg, E);
    dinv_kernel<<<(N + 255) / 256, 256, 0, stream>>>(deg, dinv, N);

    // ---- layer 0: GCNConv -> BN -> ReLU ----
    gemm_wmma_f32<128, 128, 4><<<gemmGrid128, 128, 0, stream>>>(x, W0, bufA, N);
    selfloop_bias<<<(int)((totND + 255) / 256), 256, 0, stream>>>(bufA, dinv, b0, bufB, N, D);
    scatter_edges128<<<edgeBlocks, 256, 0, stream>>>(bufA, src, dst, ew, dinv, bufB, E);
    hipMemsetAsync(colsum, 0, 128 * sizeof(float), stream);
    hipMemsetAsync(colsq,  0, 128 * sizeof(float), stream);
    bn_stats<<<statBlocks, 256, 0, stream>>>(bufB, colsum, colsq, N);
    bn_finalize<<<1, 128, 0, stream>>>(colsum, colsq, g0, be0, scsh, N);
    bn_apply_relu<<<(int)((totND + 255) / 256), 256, 0, stream>>>(bufB, scsh, totND);

    // ---- layer 1: GCNConv -> BN -> ReLU ----
    gemm_wmma_f32<128, 128, 4><<<gemmGrid128, 128, 0, stream>>>(bufB, W1, bufA, N);
    selfloop_bias<<<(int)((totND + 255) / 256), 256, 0, stream>>>(bufA, dinv, b1, bufB, N, D);
    scatter_edges128<<<edgeBlocks, 256, 0, stream>>>(bufA, src, dst, ew, dinv, bufB, E);
    hipMemsetAsync(colsum, 0, 128 * sizeof(float), stream);
    hipMemsetAsync(colsq,  0, 128 * sizeof(float), stream);
    bn_stats<<<statBlocks, 256, 0, stream>>>(bufB, colsum, colsq, N);
    bn_finalize<<<1, 128, 0, stream>>>(colsum, colsq, g1, be1, scsh, N);
    bn_apply_relu<<<(int)((totND + 255) / 256), 256, 0, stream>>>(bufB, scsh, totND);

    // ---- layer 2: GCNConv -> log_softmax ----
    gemm_wmma_f32<128, 40, 3><<<gemmGrid40, 96, 0, stream>>>(bufB, W2, out, N);
    selfloop_bias<<<(int)((totNDo + 255) / 256), 256, 0, stream>>>(out, dinv, b2, bufA, N, DOUT);
    scatter_edges40<<<edgeBlocks, 256, 0, stream>>>(out, src, dst, ew, dinv, bufA, E);
    log_softmax40<<<(N + 7) / 8, 256, 0, stream>>>(bufA, out, N);
}